// GTModeler_64072322122319
// MI455X (gfx1250) — compile-verified
//
#include <hip/hip_runtime.h>

typedef float v2f __attribute__((ext_vector_type(2)));
typedef float v8f __attribute__((ext_vector_type(8)));

__device__ __forceinline__ v8f wmma4(v2f a, v2f b, v8f c) {
  // D(16x16) = A(16x4) * B(4x16) + C, all fp32.
  // 8-arg form: (neg_a, A, neg_b, B, c_mod, C, reuse_a, reuse_b)
  return __builtin_amdgcn_wmma_f32_16x16x4_f32(false, a, false, b, (short)0, c,
                                               false, false);
}

// One block = one (batch, window). 4 waves (128 threads), wave w = head w.
// WS: window side. SH: shift. CHUNK: which 32-channel half of the input.
template <int WS, int SH, int CHUNK>
__global__ __launch_bounds__(128) void win_attn_kernel(
    const float* __restrict__ x, float* __restrict__ out) {
  constexpr int H = 256, W = 256, HW = H * W;
  constexpr int NW = W / WS;     // windows per row
  constexpr int T = WS * WS;     // tokens per window (64 or 256)
  constexpr int NT = T / 16;     // 16-token tiles (4 or 16)

  // [qv*4+head][token][ch], ch-fastest so one token's 4 channels are one row
  __shared__ float qvl[8 * T * 4];
  // per-wave 16x16 P staging tile, padded row stride 20 floats (80B: keeps
  // 8B alignment for b64 reads, bank-conflict-free for both halves)
  __shared__ float ptile[4][16 * 20];

  const int wid = blockIdx.x;
  const int b = blockIdx.y;
  const int wy = wid / NW, wx = wid % NW;
  const int tid = threadIdx.x;
  const int lane = tid & 31;
  const int head = tid >> 5;

  // ---- cooperative gather of Q (c 0..15) and V (c 16..31) into LDS ----
  const float* xbase = x + ((size_t)b * 64 + CHUNK * 32) * HW;
  for (int i = tid; i < 32 * T; i += 128) {
    int c = i / T, t = i - c * T;
    int ty = t / WS, tx = t - ty * WS;
    int gh = (wy * WS + ty - SH) & (H - 1);  // shift folded into the gather
    int gw = (wx * WS + tx - SH) & (W - 1);
    float val = xbase[c * HW + gh * W + gw];
    // c = qv*16 + head*4 + ch  ->  group g = c>>2 = qv*4+head
    qvl[((c >> 2) * T + t) * 4 + (c & 3)] = val;
  }
  __syncthreads();

  const float* qb = &qvl[head * T * 4];        // this head's Q, [token][4]
  const float* vb = &qvl[(4 + head) * T * 4];  // this head's V, [token][4]
  float* pl = &ptile[head][0];

  const int m = lane & 15;            // A-operand row / B-operand col / D col
  const int hi = lane >> 4;           // half: selects K {0,1} vs {2,3}, D row+8
  const int k0 = hi * 2;

  for (int qt = 0; qt < NT; ++qt) {
    // A operand: Q rows qt*16+m, K pair k0,k0+1 (head_dim is the K axis)
    v2f a = *(const v2f*)(qb + ((qt * 16 + m) * 4 + k0));

    // ---- scores: S[kt] = Qtile * Qtile^T  (one WMMA per key tile) ----
    v8f S[NT];
#pragma unroll
    for (int kt = 0; kt < NT; ++kt) {
      // B operand: B[k][n] = Q[kt*16+n][k] -> same LDS read pattern as A
      v2f bq = *(const v2f*)(qb + ((kt * 16 + m) * 4 + k0));
      v8f z = {0.f, 0.f, 0.f, 0.f, 0.f, 0.f, 0.f, 0.f};
      S[kt] = wmma4(a, bq, z);
    }

    // ---- row max (row M lives across one 16-lane half) + exp ----
    float rmax[8];
#pragma unroll
    for (int r = 0; r < 8; ++r) {
      float mx = -3.402823466e38f;
#pragma unroll
      for (int kt = 0; kt < NT; ++kt) mx = fmaxf(mx, S[kt][r]);
      mx = fmaxf(mx, __shfl_xor(mx, 8, 32));
      mx = fmaxf(mx, __shfl_xor(mx, 4, 32));
      mx = fmaxf(mx, __shfl_xor(mx, 2, 32));
      mx = fmaxf(mx, __shfl_xor(mx, 1, 32));
      rmax[r] = mx;
    }
#pragma unroll
    for (int kt = 0; kt < NT; ++kt) {
#pragma unroll
      for (int r = 0; r < 8; ++r) S[kt][r] = __expf(S[kt][r] - rmax[r]);
    }

    // ---- O = P * [V | 1] : stage each P tile through LDS (D->A relayout).
    // B column 4 is all-ones, so D column 4 accumulates the softmax row sum
    // inside the same WMMA chain (no ladder reduction needed). ----
    v8f O = {0.f, 0.f, 0.f, 0.f, 0.f, 0.f, 0.f, 0.f};
    for (int kt = 0; kt < NT; ++kt) {
#pragma unroll
      for (int r = 0; r < 8; ++r)
        pl[(r + 8 * hi) * 20 + m] = S[kt][r];  // (M,N) -> row-major staged
      // same-wave LDS is in-order; compiler inserts s_wait_dscnt for the reads
#pragma unroll
      for (int c4 = 0; c4 < 4; ++c4) {
        // A operand: P rows m, K = keys 4*c4 + {k0, k0+1}
        v2f pa = *(const v2f*)(pl + m * 20 + c4 * 4 + k0);
        // B operand: B[k][n] = V[kt*16 + 4*c4 + k][n] for n<4, 1.0 for n==4
        v2f bv = {0.f, 0.f};
        if (m < 4) {
          bv.x = vb[(kt * 16 + c4 * 4 + k0) * 4 + m];
          bv.y = vb[(kt * 16 + c4 * 4 + k0 + 1) * 4 + m];
        } else if (m == 4) {
          bv.x = 1.f;
          bv.y = 1.f;
        }
        O = wmma4(pa, bv, O);
      }
    }

    // ---- normalize by the WMMA-computed row sum (lane N==4) + scatter ----
#pragma unroll
    for (int r = 0; r < 8; ++r) {
      // broadcast row sum from lane N==4 of this lane's half
      float sm = __shfl(O[r], (lane & 16) | 4, 32);
      if (m < 4) {
        int tok = qt * 16 + r + 8 * hi;
        int ty = tok / WS, tx = tok - ty * WS;
        int gh = (wy * WS + ty - SH) & (H - 1);
        int gw = (wx * WS + tx - SH) & (W - 1);
        out[((size_t)b * 32 + CHUNK * 16 + head * 4 + m) * HW + gh * W + gw] =
            O[r] / sm;
      }
    }
  }
}

extern "C" void kernel_launch(void* const* d_in, const int* in_sizes, int n_in,
                              void* d_out, int out_size, void* d_ws,
                              size_t ws_size, hipStream_t stream) {
  (void)in_sizes; (void)n_in; (void)out_size; (void)d_ws; (void)ws_size;
  const float* x = (const float*)d_in[0];
  float* out = (float*)d_out;

  // chunk 0: 8x8 windows, no shift -> 32x32 windows per image
  dim3 g0(32 * 32, 4);
  win_attn_kernel<8, 0, 0><<<g0, 128, 0, stream>>>(x, out);
  // chunk 1: 16x16 windows, shift (8,8) -> 16x16 windows per image
  dim3 g1(16 * 16, 4);
  win_attn_kernel<16, 8, 1><<<g1, 128, 0, stream>>>(x, out);
}